// DiagonalTraining_82085414961327
// MI455X (gfx1250) — compile-verified
//
#include <hip/hip_runtime.h>
#include <cstdint>

// Problem constants (reference: B=32, S=512)
#define SDIM 512
#define BDIM 32
#define KC 32                      // K-chunk (r) per TDM stage
#define NCHUNK (SDIM / KC)         // 16
#define THREADS 256                // 8 waves (wave32)

// LDS layout (floats). TDM pad features give conflict-free strides.
#define D_STRIDE 516               // 512 + 2 pad dwords per 256 (TDM pad_interval=256, pad=2)
#define D_FLOATS (BDIM * D_STRIDE) // 16512
#define W_ROW 34                   // 32 + 2 pad dwords per 32  (TDM pad_interval=32,  pad=2)
#define W_FLOATS (SDIM * W_ROW)    // 17408 per buffer
#define SMEM_FLOATS (D_FLOATS + 2 * W_FLOATS)

typedef float v2f __attribute__((ext_vector_type(2)));
typedef float v8f __attribute__((ext_vector_type(8)));
typedef unsigned int v4u __attribute__((ext_vector_type(4)));
typedef int v4i __attribute__((ext_vector_type(4)));
typedef int v8i __attribute__((ext_vector_type(8)));

#if defined(__AMDGCN__) && __has_builtin(__builtin_amdgcn_tensor_load_to_lds)
#define HAVE_TDM 1
#else
#define HAVE_TDM 0
#endif

#if HAVE_TDM
__device__ __forceinline__ void tdm_issue(v4u g0, v8i g1, v4i g2, v4i g3) {
#if __clang_major__ >= 23
  v8i g4 = {0, 0, 0, 0, 0, 0, 0, 0};
  __builtin_amdgcn_tensor_load_to_lds(g0, g1, g2, g3, g4, 0);
#else
  __builtin_amdgcn_tensor_load_to_lds(g0, g1, g2, g3, 0);
#endif
}

// D# group0: count=1, lds_addr, 57-bit global addr, type=2 ("image")
__device__ __forceinline__ v4u make_g0(uint32_t lds_addr, uint64_t gaddr) {
  v4u g0;
  g0.x = 1u;
  g0.y = lds_addr;
  g0.z = (uint32_t)(gaddr & 0xFFFFFFFFu);
  g0.w = (uint32_t)((gaddr >> 32) & 0x1FFFFFFu) | (2u << 30);
  return g0;
}

// D# group1: data_size=4B, pad_enable, pad_interval/amount codes, dims/strides in elements
__device__ __forceinline__ v8i make_g1(uint32_t td0, uint32_t td1,
                                       uint32_t tile0, uint32_t tile1, uint32_t tile2,
                                       uint64_t s0, uint64_t s1,
                                       uint32_t padI, uint32_t padA) {
  v8i g;
  g[0] = (int)((2u << 16) | (1u << 20) | (padI << 22) | (padA << 25));
  g[1] = (int)((td0 & 0xFFFFu) << 16);
  g[2] = (int)((td0 >> 16) | ((td1 & 0xFFFFu) << 16));
  g[3] = (int)((td1 >> 16) | (tile0 << 16));
  g[4] = (int)(tile1 | (tile2 << 16));
  g[5] = (int)(s0 & 0xFFFFFFFFu);
  g[6] = (int)(((s0 >> 32) & 0xFFFFu) | ((s1 & 0xFFFFu) << 16));
  g[7] = (int)((s1 >> 16) & 0xFFFFFFFFu);
  return g;
}

// Stage one K-chunk of W[i] (512 q-rows x 32 r-cols) into LDS buffer `buf`.
__device__ __forceinline__ void issue_w(uint32_t ldsBase, const float* W, int i, int kc, int buf) {
  uint64_t wg = (uint64_t)(uintptr_t)(W + (size_t)i * SDIM * SDIM + (size_t)kc * KC);
  v4u g0 = make_g0(ldsBase + (uint32_t)(D_FLOATS + buf * W_FLOATS) * 4u, wg);
  // 2D: tile 32x512 of a 512x512 tensor, row stride 512; pad 2dw/32dw -> LDS row stride 34
  v8i g1 = make_g1(SDIM, SDIM, KC, SDIM, 0, SDIM, 0, /*padI(32dw)*/4, /*padA(2dw)*/1);
  v4i g2 = {0, 0, 0, 0};
  v4i g3 = {0, 0, 0, 0};
  tdm_issue(g0, g1, g2, g3);
}
#endif  // HAVE_TDM

// Upper anti-triangle passthrough: y[b,i,j] = x[b,i,j] where i+j >= S.
__global__ __launch_bounds__(256) void diag_copy_kernel(const float* __restrict__ x,
                                                        float* __restrict__ y) {
  size_t idx = (size_t)blockIdx.x * 256u + threadIdx.x;
  int j = (int)(idx & (SDIM - 1));
  int ii = (int)((idx >> 9) & (SDIM - 1));
  if (ii + j >= SDIM) y[idx] = x[idx];
}

// One block per i: out_i(32x512) = D_i(32x512) @ W[i]^T(512x512) + bias[i], scattered to y.
__global__ __launch_bounds__(THREADS) void diag_gemm_kernel(const float* __restrict__ x,
                                                            const float* __restrict__ W,
                                                            const float* __restrict__ bias,
                                                            float* __restrict__ y) {
  extern __shared__ float smem[];
  float* sD = smem;
  float* sW0 = smem + D_FLOATS;
  float* sW1 = sW0 + W_FLOATS;

  const int i = blockIdx.x;
  const int tid = (int)threadIdx.x;
  const int lane = tid & 31;
  const int wave = tid >> 5;
  const int mrow = lane & 15;
  const int half = lane >> 4;

  v8f c[2][4] = {};  // 2 M-tiles (batch 0-15 / 16-31) x 4 N-tiles (q) per wave

#if HAVE_TDM
  const uint32_t ldsBase = (uint32_t)(uintptr_t)smem;  // LDS offset (low 32b of aspace(3) ptr)
  if (wave == 0) {
    // D_i gather: 3D descriptor. dim0 = 1 element; dim1 = r with elem stride S-1
    // (x[b,r,i-r] = x + i + r*(S-1)); dim2 = b with stride S*S.
    // tensor_dim1 = i+1 -> TDM OOB zero-fill implements the causal mask.
    uint64_t xg = (uint64_t)(uintptr_t)(x + i);
    v4u g0 = make_g0(ldsBase, xg);
    v8i g1 = make_g1(/*td0*/1, /*td1*/(uint32_t)(i + 1), /*tile0*/1, /*tile1*/SDIM,
                     /*tile2*/BDIM, /*s0*/(uint64_t)(SDIM - 1), /*s1*/(uint64_t)SDIM * SDIM,
                     /*padI(256dw)*/7, /*padA(2dw)*/1);
    v4i g2 = {BDIM, 0, 0, 0};  // tensor_dim2 = 32
    v4i g3 = {0, 0, 0, 0};
    tdm_issue(g0, g1, g2, g3);
    issue_w(ldsBase, W, i, 0, 0);
  }
#endif

  for (int kc = 0; kc < NCHUNK; ++kc) {
#if HAVE_TDM
    if (wave == 0) {
      if (kc + 1 < NCHUNK) {
        issue_w(ldsBase, W, i, kc + 1, (kc + 1) & 1);   // prefetch next chunk
        __builtin_amdgcn_s_wait_tensorcnt(1);           // D + W[kc] complete (in-order)
      } else {
        __builtin_amdgcn_s_wait_tensorcnt(0);
      }
    }
    __syncthreads();
#else
    if (kc == 0) {
      for (int t = tid; t < BDIM * SDIM; t += THREADS) {
        int b_ = t >> 9, r = t & (SDIM - 1);
        float v_ = (r <= i) ? x[((size_t)b_ * SDIM + r) * SDIM + (i - r)] : 0.0f;
        sD[b_ * D_STRIDE + r + ((r >= 256) ? 2 : 0)] = v_;
      }
    }
    {
      float* dst = (kc & 1) ? sW1 : sW0;
      const float* src = W + (size_t)i * SDIM * SDIM + (size_t)kc * KC;
      for (int t = tid; t < SDIM * KC; t += THREADS) {
        int q = t >> 5, cc2 = t & 31;
        dst[q * W_ROW + cc2] = src[(size_t)q * SDIM + cc2];
      }
    }
    __syncthreads();
#endif

    const float* Wb = (kc & 1) ? sW1 : sW0;
#pragma unroll
    for (int ks = 0; ks < KC / 4; ++ks) {
      // f32 A operand (16x4): lane m holds K={k,k+1} (lanes<16) / K={k+2,k+3} (lanes>=16)
      const int kk = ks * 4 + 2 * half;
      const int r = kc * KC + kk;
      const int radj = r + ((r >= 256) ? 2 : 0);
      v2f a0 = *(const v2f*)(sD + mrow * D_STRIDE + radj);
      v2f a1 = *(const v2f*)(sD + (mrow + 16) * D_STRIDE + radj);
#pragma unroll
      for (int j = 0; j < 4; ++j) {
        // f32 B operand (4x16): lane n=mrow holds B[k..k+1,n] = W[q0+n, r..r+1]
        const int q = (wave * 4 + j) * 16 + mrow;
        v2f bm = *(const v2f*)(Wb + q * W_ROW + kk);
        c[0][j] = __builtin_amdgcn_wmma_f32_16x16x4_f32(false, a0, false, bm, (short)0,
                                                        c[0][j], false, false);
        c[1][j] = __builtin_amdgcn_wmma_f32_16x16x4_f32(false, a1, false, bm, (short)0,
                                                        c[1][j], false, false);
      }
    }
    __syncthreads();  // everyone done with buf[kc&1] before it is refilled at kc+2
  }

  // Epilogue: bias add + diagonal scatter  y[b, q, i-q] = out[b,i,q]  (q <= i)
#pragma unroll
  for (int mt = 0; mt < 2; ++mt) {
#pragma unroll
    for (int j = 0; j < 4; ++j) {
      const int q = (wave * 4 + j) * 16 + mrow;
      const float bv = bias[i * SDIM + q];
      if (q <= i) {
        const int col = i - q;
#pragma unroll
        for (int v = 0; v < 8; ++v) {
          const int b_ = mt * 16 + v + 8 * half;  // C layout: VGPR v, lane half
          y[((size_t)b_ * SDIM + q) * SDIM + col] = c[mt][j][v] + bv;
        }
      }
    }
  }
}

extern "C" void kernel_launch(void* const* d_in, const int* in_sizes, int n_in,
                              void* d_out, int out_size, void* d_ws, size_t ws_size,
                              hipStream_t stream) {
  (void)in_sizes; (void)n_in; (void)out_size; (void)d_ws; (void)ws_size;
  const float* x = (const float*)d_in[0];
  const float* W = (const float*)d_in[1];
  const float* b = (const float*)d_in[2];
  float* y = (float*)d_out;

  // 1) upper anti-triangle keeps x (disjoint from the scatter region)
  diag_copy_kernel<<<(BDIM * SDIM * SDIM) / 256, 256, 0, stream>>>(x, y);
  // 2) 512 blocks, one per i; ~200KB dynamic LDS (CDNA5 WGP has 320KB)
  diag_gemm_kernel<<<SDIM, THREADS, SMEM_FLOATS * sizeof(float), stream>>>(x, W, b, y);
}